// flow_mapping_20031727468850
// MI455X (gfx1250) — compile-verified
//
#include <hip/hip_runtime.h>

#define BOUND     10.0f
#define BETA      1e-6f
#define LOG2_BETA -19.931568569324174f      /* log2(1e-6) */
#define LN2       0.6931471805599453f
#define C1        1.74884258895036416f      /* (1.2^16 - 1)/10  */
#define INV_L2R   3.801784016923931f        /* 1/log2(1.2)      */
#define X1L       0.114361452442796272f     /* 2/(1.2^16 - 1)   */

typedef float f32x4 __attribute__((ext_vector_type(4)));

// Tables in LDS:
//  me_s[k] = (mesh[k], 1/elmt[k])            k = 0..32
//  pf_s[k*9+d] = (pdf[k][d], F_ref[k][d])    stride 9 -> conflict-free b64 gathers
__global__ __launch_bounds__(256) void flow_mapping_kernel(
    const float* __restrict__ x,
    const float* __restrict__ logdet_in,
    const float* __restrict__ p,
    float* __restrict__ out,
    float* __restrict__ ldout,
    long long nrows)
{
    __shared__ float2 me_s[33];
    __shared__ float  elmt_s[33];
    __shared__ float2 pf_s[33 * 9];

    const int t = threadIdx.x;

    // ---- build mesh ----
    if (t < 33) {
        float idx = (float)t - 16.0f;
        float a   = fabsf(idx);
        float xr  = (1.0f - powf(1.2f, a)) / (1.0f - 1.2f);
        xr = (idx >= 0.0f) ? X1L * xr : -X1L * xr;
        xr = (xr + BOUND) / (2.0f * BOUND);
        me_s[t].x = (t == 0) ? 0.0f : ((t == 32) ? 1.0f : xr);
    }
    __syncthreads();
    if (t < 32) {
        float h = me_s[t + 1].x - me_s[t].x;
        elmt_s[t] = h;
        me_s[t].y = 1.0f / h;
    } else if (t == 32) {
        elmt_s[32] = 0.0f;
        me_s[32].y = 0.0f;
    }
    __syncthreads();
    // ---- build pdf / F_ref columns (one lane per dim) ----
    if (t < 8) {
        float denom = 0.0f;
        for (int j = 0; j < 31; ++j) {
            float w = 0.5f * (elmt_s[j] + elmt_s[j + 1]);
            denom += expf(p[j * 8 + t]) * w;
        }
        float norm = (1.0f - (elmt_s[0] + elmt_s[31]) * (0.5f * BETA)) / denom;
        pf_s[0 * 9 + t].x = BETA;
        for (int j = 0; j < 31; ++j)
            pf_s[(j + 1) * 9 + t].x = norm * expf(p[j * 8 + t]);
        pf_s[32 * 9 + t].x = BETA;
        float F = 0.0f;
        for (int k = 0; k < 32; ++k) {
            pf_s[k * 9 + t].y = F;
            F += 0.5f * (pf_s[k * 9 + t].x + pf_s[(k + 1) * 9 + t].x) * elmt_s[k];
        }
        pf_s[32 * 9 + t].y = F;
    }
    __syncthreads();

    // ---- streaming main loop: one row (8 dims) per thread, grid-stride ----
    const long long stride = (long long)gridDim.x * blockDim.x;
    for (long long row = (long long)blockIdx.x * blockDim.x + t; row < nrows; row += stride) {
        const f32x4* xin = (const f32x4*)(x + (size_t)row * 8);
        f32x4 a0 = __builtin_nontemporal_load(xin);
        f32x4 a1 = __builtin_nontemporal_load(xin + 1);
        float xv[8] = {a0.x, a0.y, a0.z, a0.w, a1.x, a1.y, a1.z, a1.w};
        float yv[8];
        float dld2 = 0.0f;   // accumulated in log2 space, scaled by ln2 at the end

#pragma unroll
        for (int d = 0; d < 8; ++d) {
            float s  = xv[d];
            float tB = s + BOUND;                       // x + 10
            // cover == (0 <= (x+10)/20 < 1), exact for all f32 without dividing
            bool cover = (tB >= 0.0f) && (tB < 2.0f * BOUND);
            float u  = tB * (0.5f / BOUND);

            // analytic geometric-mesh bin index (replaces binary search)
            float sa  = fabsf(s);
            float lg  = __log2f(fmaf(sa, C1, 1.0f)) * INV_L2R;
            int   i   = (int)lg;                        // lg >= 0, trunc == floor
            int   k   = (s >= 0.0f) ? (16 + i) : (15 - i);
            k = min(max(k, 0), 31);

            float2 me  = me_s[k];                       // (mesh[k], 1/h)
            float2 pf1 = pf_s[k * 9 + d];               // (v1, Fp)
            float  v2  = pf_s[k * 9 + 9 + d].x;         // pdf[k+1][d]

            float xm    = u - me.x;
            float slope = (v2 - pf1.x) * me.y;

            float y  = cover ? (pf1.y + (0.5f * xm * xm) * slope + xm * pf1.x) : u;
            float tt = cover ? (xm * slope + pf1.x) : 1.0f;
            dld2 += __log2f(tt);                        // native v_log_f32

            float o = y * (2.0f * BOUND) - BOUND;
            if (o > BOUND)  { o = BETA * (o - BOUND) + BOUND; dld2 += LOG2_BETA; }
            if (o < -BOUND) { o = BETA * (o + BOUND) - BOUND; dld2 += LOG2_BETA; }
            yv[d] = o;
        }

        f32x4* op = (f32x4*)(out + (size_t)row * 8);
        f32x4 o0 = {yv[0], yv[1], yv[2], yv[3]};
        f32x4 o1 = {yv[4], yv[5], yv[6], yv[7]};
        __builtin_nontemporal_store(o0, op);
        __builtin_nontemporal_store(o1, op + 1);

        float ld = __builtin_nontemporal_load(logdet_in + row) + dld2 * LN2;
        __builtin_nontemporal_store(ld, ldout + row);
    }
}

extern "C" void kernel_launch(void* const* d_in, const int* in_sizes, int n_in,
                              void* d_out, int out_size, void* d_ws, size_t ws_size,
                              hipStream_t stream) {
    (void)n_in; (void)out_size; (void)d_ws; (void)ws_size;
    const float* x      = (const float*)d_in[0];
    const float* logdet = (const float*)d_in[1];
    const float* p      = (const float*)d_in[2];

    long long nrows = (long long)in_sizes[0] / 8;
    float* out   = (float*)d_out;
    float* ldout = out + (size_t)nrows * 8;

    const int threads = 256;
    long long need = (nrows + threads - 1) / threads;
    int blocks = (int)((need < 4096) ? need : 4096);
    if (blocks < 1) blocks = 1;

    hipLaunchKernelGGL(flow_mapping_kernel, dim3(blocks), dim3(threads), 0, stream,
                       x, logdet, p, out, ldout, nrows);
}